// periodic_padding_59313498357812
// MI455X (gfx1250) — compile-verified
//
#include <hip/hip_runtime.h>
#include <stddef.h>

// Circular padding gather: out[b,c,yo,xo] = x[b,c,(yo-ELP) mod N, (xo-ELP) mod N]
// x: [8,16,512,512] f32  ->  out: [8,16,520,520] f32.
// Pure data movement (0 FLOPs, ~260 MB traffic => ~11us at 23.3 TB/s).
// gfx1250 path: async global->LDS copies (ASYNCcnt) stage each wrapped row in
// LDS, then aligned b128 stores emit the output row fully coalesced.

#define ELP 4
#define ERP 4
#define N   512
#define NO  (N + ELP + ERP)   // 520
#define BLOCK 256

#if defined(__has_builtin)
#  if __has_builtin(__builtin_amdgcn_global_load_async_to_lds_b128) && \
      __has_builtin(__builtin_amdgcn_global_load_async_to_lds_b32)  && \
      __has_builtin(__builtin_amdgcn_s_wait_asynccnt)
#    define USE_ASYNC_LDS 1
#  endif
#endif
#ifndef USE_ASYNC_LDS
#  define USE_ASYNC_LDS 0
#endif

typedef __attribute__((ext_vector_type(4))) float f32x4;

#if USE_ASYNC_LDS
// Builtin parameter types per hipcc diagnostic:
//   b128: (v4i addrspace(1)*, v4i addrspace(3)*, imm, imm)
//   b32 : (int addrspace(1)*, int addrspace(3)*, imm, imm)
typedef int v4i __attribute__((vector_size(16)));
typedef __attribute__((address_space(1))) v4i* g_v4i;
typedef __attribute__((address_space(3))) v4i* l_v4i;
typedef __attribute__((address_space(1))) int* g_i32;
typedef __attribute__((address_space(3))) int* l_i32;
#endif

__global__ __launch_bounds__(BLOCK) void periodic_pad_rows(
    const float* __restrict__ x, float* __restrict__ out) {
  // Padded source row staged in LDS: row[i] = src[(i - ELP) mod N], i in [0,NO)
  __shared__ __align__(16) float row[NO];

  const int      t  = (int)threadIdx.x;
  const unsigned id = blockIdx.x;
  const unsigned yo = id % NO;          // output row within image
  const unsigned bc = id / NO;          // fused batch*channel index
  const unsigned ysrc = (yo + (N - ELP)) % N;

  const float* __restrict__ src = x   + ((size_t)bc * N  + ysrc) * N;   // 2048B-aligned rows
  float*       __restrict__ dst = out + ((size_t)bc * NO + yo)   * NO;  // 2080B stride, 16B-aligned

#if USE_ASYNC_LDS
  if (t < N / 4) {
    // Interior: 128 lanes x 16B async copies (global src and LDS dst both 16B-aligned).
    __builtin_amdgcn_global_load_async_to_lds_b128(
        (g_v4i)(src + 4 * t), (l_v4i)(&row[ELP + 4 * t]), 0, 0);
  } else if (t < N / 4 + ELP) {
    // Left wrap: row[0..ELP) = src[N-ELP .. N)
    const int j = t - N / 4;
    __builtin_amdgcn_global_load_async_to_lds_b32(
        (g_i32)(src + (N - ELP) + j), (l_i32)(&row[j]), 0, 0);
  } else if (t < N / 4 + ELP + ERP) {
    // Right wrap: row[ELP+N .. NO) = src[0..ERP)
    const int j = t - (N / 4 + ELP);
    __builtin_amdgcn_global_load_async_to_lds_b32(
        (g_i32)(src + j), (l_i32)(&row[ELP + N + j]), 0, 0);
  }
  // Our wave's async copies must have landed in LDS before the barrier
  // releases other waves to read them.
  __builtin_amdgcn_s_wait_asynccnt(0);
#else
  // Fallback: plain global->reg->LDS staging.
  for (int i = t; i < NO; i += BLOCK) {
    const int xs = (i + (N - ELP)) % N;
    row[i] = src[xs];
  }
#endif
  __syncthreads();

  // Emit the padded row as 130 aligned 16B stores (fully coalesced).
  if (t < NO / 4) {
    *(f32x4*)(dst + 4 * t) = *(const f32x4*)(&row[4 * t]);
  }
}

extern "C" void kernel_launch(void* const* d_in, const int* in_sizes, int n_in,
                              void* d_out, int out_size, void* d_ws, size_t ws_size,
                              hipStream_t stream) {
  (void)n_in; (void)out_size; (void)d_ws; (void)ws_size;
  const float* x = (const float*)d_in[0];
  float* out = (float*)d_out;
  // elp/erp are compile-time 4 (setup_inputs); shapes are static.
  const int bc = in_sizes[0] / (N * N);       // 8*16 = 128
  const unsigned nblocks = (unsigned)bc * NO; // 128 * 520 = 66560
  periodic_pad_rows<<<nblocks, BLOCK, 0, stream>>>(x, out);
}